// LeakyRNN_72421738545198
// MI455X (gfx1250) — compile-verified
//
#include <hip/hip_runtime.h>
#include <math.h>

// Problem constants (from reference)
#define T_STEPS 64
#define B_SZ    16
#define IN_SZ   128
#define H_SZ    256
#define OUT_SZ  32
#define G_SZ    4
#define E_SIZE  204   // int(0.8*256)    -> ei = +1 for j<204, -1 otherwise
#define ROW_ONES 205  // H - int((1-0.8)*256) = 256 - 51 -> mask_row ones count
#define MASK_BULK 204 // largest multiple of 4 such that k0+3 < ROW_ONES for all k0 below
#define ALPHA_X 0.2f
#define OMAX    0.8f
#define ALPHA_W 0.1f
#define OMAW    0.9f

// LDS row strides padded so bank = f(row) -> conflict-free one-row-per-lane access
#define WX_STRIDE (IN_SZ + 1)   // 129 floats  (129 dwords)
#define WH_STRIDE (H_SZ + 2)    // 258 bf16    (129 dwords)

static __device__ __forceinline__ float bf2f(unsigned short u) {
  return __uint_as_float(((unsigned int)u) << 16);
}
static __device__ __forceinline__ unsigned short f2bf(float f) {
  unsigned int u = __float_as_uint(f);
  u += 0x7FFFu + ((u >> 16) & 1u);           // round-to-nearest-even
  return (unsigned short)(u >> 16);
}

// ---------------------------------------------------------------------------
// Kernel 1: whole recurrence for one sample per workgroup.
// Plastic weights stay resident in LDS for all 64 timesteps:
//   wx: fp32 [H][129]  (128 KB + pad)     wh: bf16 [H][258]  (128 KB + pad)
// 256 threads (8 wave32s): thread t owns output row o=t of both matrices.
// Avoids 64 x 12 MB of HBM round-trips for the plastic state (~768 MB).
// ---------------------------------------------------------------------------
__global__ void __launch_bounds__(256)
rnn_plastic_resident_kernel(const float* __restrict__ x,   // [T,B,IN]
                            const float* __restrict__ Rs,  // [T]
                            const float* __restrict__ Wx,  // [H,IN]
                            const float* __restrict__ bx,  // [H]
                            const float* __restrict__ Wh,  // [H,H]
                            const float* __restrict__ bh,  // [H]
                            const float* __restrict__ Wa,  // [G,H]
                            const float* __restrict__ ba,  // [G]
                            const float* __restrict__ cp,  // [6]
                            float* __restrict__ hs)        // [T,B,H]
{
  extern __shared__ unsigned char smem[];
  float*          s_wx   = (float*)smem;                                // H*WX_STRIDE f32
  unsigned short* s_wh   = (unsigned short*)(s_wx + H_SZ * WX_STRIDE);  // H*WH_STRIDE bf16
  float*          s_outA = (float*)(s_wh + H_SZ * WH_STRIDE);           // H
  float*          s_outB = s_outA + H_SZ;                               // H
  float*          s_xt   = s_outB + H_SZ;                               // IN
  float*          s_red  = s_xt + IN_SZ;                                // 4*8 wave partials
  float*          s_aw   = s_red + 4 * 8;                               // G

  const int tid  = threadIdx.x;    // 0..255 == output row o
  const int wave = tid >> 5;       // 0..7
  const int lane = tid & 31;
  const int b    = blockIdx.x;     // sample

  // ---- init resident state ------------------------------------------------
  for (int i = 0; i < IN_SZ; ++i) s_wx[tid * WX_STRIDE + i] = Wx[tid * IN_SZ + i];
  for (int j = 0; j < H_SZ;  ++j) s_wh[tid * WH_STRIDE + j] = f2bf(Wh[tid * H_SZ + j]);
  s_outA[tid] = 0.0f;
  s_outB[tid] = 0.0f;

  float state = 0.0f;                                  // per-row leaky state in a register
  const float c0 = fabsf(cp[0]), c1 = fabsf(cp[1]), c2 = fabsf(cp[2]);
  const float c3 = fabsf(cp[3]), c4 = fabsf(cp[4]), c5 = fabsf(cp[5]);
  const float bias_o = bx[tid] + bh[tid];
  const float rowm = (tid < ROW_ONES) ? 1.0f : 0.0f;   // mask_row over H (attention input)
  const float wa0 = fabsf(Wa[0 * H_SZ + tid]) * rowm;
  const float wa1 = fabsf(Wa[1 * H_SZ + tid]) * rowm;
  const float wa2 = fabsf(Wa[2 * H_SZ + tid]) * rowm;
  const float wa3 = fabsf(Wa[3 * H_SZ + tid]) * rowm;
  __syncthreads();

  float* s_out  = s_outA;   // t-th output (read)
  float* s_nout = s_outB;   // (t+1)-th output (write)

  for (int t = 0; t < T_STEPS; ++t) {
    // prefetch next timestep's x slice into cache (global_prefetch_b8)
    if (tid < IN_SZ && (t + 1) < T_STEPS)
      __builtin_prefetch(&x[((size_t)(t + 1) * B_SZ + b) * IN_SZ + tid], 0, 1);

    const float R = Rs[t];

    // ---- attention logits: in-wave shuffle butterfly, then 8-way combine --
    const float ov = s_out[tid];
    float p0 = ov * wa0, p1 = ov * wa1, p2 = ov * wa2, p3 = ov * wa3;
#pragma unroll
    for (int m = 16; m >= 1; m >>= 1) {   // wave32 butterfly, no barriers
      p0 += __shfl_xor(p0, m);
      p1 += __shfl_xor(p1, m);
      p2 += __shfl_xor(p2, m);
      p3 += __shfl_xor(p3, m);
    }
    if (lane == 0) {
      s_red[0 * 8 + wave] = p0;
      s_red[1 * 8 + wave] = p1;
      s_red[2 * 8 + wave] = p2;
      s_red[3 * 8 + wave] = p3;
    }
    __syncthreads();
    if (tid == 0) {  // combine 8 wave partials + softmax over G=4
      float l0 = ba[0], l1 = ba[1], l2 = ba[2], l3 = ba[3];
#pragma unroll
      for (int w = 0; w < 8; ++w) {
        l0 += s_red[w]; l1 += s_red[8 + w]; l2 += s_red[16 + w]; l3 += s_red[24 + w];
      }
      float mx = fmaxf(fmaxf(l0, l1), fmaxf(l2, l3));
      float e0 = expf(l0 - mx), e1 = expf(l1 - mx), e2 = expf(l2 - mx), e3 = expf(l3 - mx);
      float inv = 1.0f / (e0 + e1 + e2 + e3);
      s_aw[0] = e0 * inv; s_aw[1] = e1 * inv; s_aw[2] = e2 * inv; s_aw[3] = e3 * inv;
    }
    __syncthreads();

    // ---- gated input xt ---------------------------------------------------
    if (tid < IN_SZ)
      s_xt[tid] = x[((size_t)t * B_SZ + b) * IN_SZ + tid] * s_aw[tid >> 5];
    __syncthreads();

    // ---- mat-vec over this thread's private rows --------------------------
    float acc = bias_o;
#pragma unroll 8
    for (int i = 0; i < IN_SZ; ++i)
      acc = fmaf(fabsf(s_wx[tid * WX_STRIDE + i]), s_xt[i], acc);   // |wx| (PosWLinear)
#pragma unroll 8
    for (int j = 0; j < H_SZ; ++j) {
      float w   = bf2f(s_wh[tid * WH_STRIDE + j]);
      float eff = fmaxf(w, 0.0f) * ((j < E_SIZE) ? 1.0f : -1.0f);   // relu * ei
      eff       = (j == tid) ? 0.0f : eff;                          // zero diagonal
      acc = fmaf(eff, s_out[j], acc);
    }
    state = state * OMAX + acc * ALPHA_X;
    const float no = tanhf(fmaxf(state, 0.0f));                     // retanh
    s_nout[tid] = no;
    hs[((size_t)t * B_SZ + b) * H_SZ + tid] = no;

    // ---- Hebbian plasticity on private rows (no barrier needed: row-private)
    const float awr = ALPHA_W * R;
    const float kA = awr * (c0 + c2 * no);   // (c0 + c2*nout)*xt
    const float kB = awr * c1 * no;          //  c1*nout
#pragma unroll 8
    for (int i = 0; i < IN_SZ; ++i) {
      float w = s_wx[tid * WX_STRIDE + i];
      s_wx[tid * WX_STRIDE + i] = fmaf(kA, s_xt[i], w * OMAW) + kB;
    }
    const float kC = awr * (c3 + c5 * no);
    const float kD = awr * c4 * no;
#pragma unroll 8
    for (int j = 0; j < H_SZ; ++j) {
      float w = bf2f(s_wh[tid * WH_STRIDE + j]);
      s_wh[tid * WH_STRIDE + j] = f2bf(fmaf(kC, s_out[j], w * OMAW) + kD);
    }
    __syncthreads();                 // everyone done reading s_out / s_xt
    float* tmp = s_out; s_out = s_nout; s_nout = tmp;
  }
}

// ---------------------------------------------------------------------------
// Kernel 2: output projection  out[1024,32] = hs[1024,256] @ (|Wo|·mask_row)ᵀ + bo
// Full-fp32 WMMA: V_WMMA_F32_16X16X4_F32, one wave per 16x16 tile, K-chain of 64.
// K-loop split: bulk [0,204) where mask_row==1 for every lane (body is just
// 2x global_load_b64 + 2x v_and (|Wo|) + v_wmma), masked tail [204,256).
// Lane layouts per CDNA5 ISA §7.12.2:
//   A 16x4 f32 : lanes 0-15 hold M=lane, K={0,1}; lanes 16-31 hold M=lane-16, K={2,3}
//   B 4x16 f32 : lanes 0-15 hold N=lane, K={0,1}; lanes 16-31 hold N=lane-16, K={2,3}
//   C/D 16x16  : VGPR r -> M=r (lanes 0-15) / M=8+r (lanes 16-31), N=lane&15
// ---------------------------------------------------------------------------
typedef float v2f __attribute__((ext_vector_type(2)));
typedef float v8f __attribute__((ext_vector_type(8)));

__global__ void __launch_bounds__(32)
proj_wmma_f32_kernel(const float* __restrict__ hs,  // [T*B, H]
                     const float* __restrict__ Wo,  // [OUT, H]
                     const float* __restrict__ bo,  // [OUT]
                     float* __restrict__ out)       // [T*B, OUT]
{
  const int lane = threadIdx.x;     // 0..31, full wave, EXEC all 1s throughout
  const int half = lane >> 4;
  const int l    = lane & 15;
  const int mt   = blockIdx.x >> 1;       // 64 M-tiles of 16 rows
  const int nt   = blockIdx.x & 1;        // 2  N-tiles of 16 cols
  const int m    = mt * 16 + l;           // A row held by this lane
  const int n    = nt * 16 + l;           // B col held by this lane

  const float* hrow = hs + (size_t)m * H_SZ;
  const float* wrow = Wo + (size_t)n * H_SZ;

  const float bias = bo[n];
  v8f c;
#pragma unroll
  for (int r = 0; r < 8; ++r) c[r] = bias;   // C pre-loaded with bo (broadcast over M)

  // ---- bulk: mask_row == 1 for all lanes (k0+3 < ROW_ONES) -----------------
#pragma unroll 8
  for (int k0 = 0; k0 < MASK_BULK; k0 += 4) {   // 51 iterations
    const int ka = k0 + half * 2;               // lanes 16-31 take K={2,3}
    v2f a, bb;
    a.x  = hrow[ka];
    a.y  = hrow[ka + 1];
    bb.x = fabsf(wrow[ka]);                     // Wo_eff = |Wo| (mask == 1 here)
    bb.y = fabsf(wrow[ka + 1]);
    c = __builtin_amdgcn_wmma_f32_16x16x4_f32(
        false, a, false, bb, (short)0, c, false, false);
  }

  // ---- tail: apply mask_row as a 0/1 multiplicand (branch-free) ------------
#pragma unroll
  for (int k0 = MASK_BULK; k0 < H_SZ; k0 += 4) { // 13 iterations
    const int ka = k0 + half * 2;
    const float mkx = (ka     < ROW_ONES) ? 1.0f : 0.0f;
    const float mky = (ka + 1 < ROW_ONES) ? 1.0f : 0.0f;
    v2f a, bb;
    a.x  = hrow[ka];
    a.y  = hrow[ka + 1];
    bb.x = fabsf(wrow[ka])     * mkx;
    bb.y = fabsf(wrow[ka + 1]) * mky;
    c = __builtin_amdgcn_wmma_f32_16x16x4_f32(
        false, a, false, bb, (short)0, c, false, false);
  }

#pragma unroll
  for (int r = 0; r < 8; ++r)
    out[(size_t)(mt * 16 + half * 8 + r) * OUT_SZ + n] = c[r];
}

// ---------------------------------------------------------------------------
extern "C" void kernel_launch(void* const* d_in, const int* in_sizes, int n_in,
                              void* d_out, int out_size, void* d_ws, size_t ws_size,
                              hipStream_t stream) {
  const float* x  = (const float*)d_in[0];
  const float* Rs = (const float*)d_in[1];
  const float* Wx = (const float*)d_in[2];
  const float* bx = (const float*)d_in[3];
  const float* Wh = (const float*)d_in[4];
  const float* bh = (const float*)d_in[5];
  const float* Wo = (const float*)d_in[6];
  const float* bo = (const float*)d_in[7];
  const float* Wa = (const float*)d_in[8];
  const float* ba = (const float*)d_in[9];
  const float* cp = (const float*)d_in[10];

  float* out = (float*)d_out;                                  // [T,B,OUT]
  float* hs  = out + (size_t)T_STEPS * B_SZ * OUT_SZ;          // [T,B,H]

  // dynamic LDS: wx(fp32,padded) + wh(bf16,padded) + out double-buffer + xt + red + aw
  const size_t smem = (size_t)H_SZ * WX_STRIDE * sizeof(float)
                    + (size_t)H_SZ * WH_STRIDE * sizeof(unsigned short)
                    + (size_t)(2 * H_SZ + IN_SZ + 4 * 8 + G_SZ) * sizeof(float);
  // ~262 KB — fits CDNA5's 320 KB-per-workgroup LDS; one workgroup per WGP.

  rnn_plastic_resident_kernel<<<B_SZ, 256, smem, stream>>>(
      x, Rs, Wx, bx, Wh, bh, Wa, ba, cp, hs);

  const int n_tiles = (T_STEPS * B_SZ / 16) * (OUT_SZ / 16);   // 64 * 2 = 128
  proj_wmma_f32_kernel<<<n_tiles, 32, 0, stream>>>(hs, Wo, bo, out);
}